// GAT_72919954751657
// MI455X (gfx1250) — compile-verified
//
#include <hip/hip_runtime.h>
#include <hip/hip_bf16.h>

#define HIDDEN 128
#define LEAKY 0.01f

typedef __attribute__((ext_vector_type(2))) float v2f;
typedef __attribute__((ext_vector_type(8))) float v8f;

// ---------------------------------------------------------------------------
// order-preserving float <-> uint key (for atomicMax-based segment max)
// ---------------------------------------------------------------------------
__device__ __forceinline__ unsigned f32_key(float f) {
  unsigned u = __float_as_uint(f);
  return (u & 0x80000000u) ? ~u : (u | 0x80000000u);
}
__device__ __forceinline__ float key_f32(unsigned k) {
  unsigned u = (k & 0x80000000u) ? (k ^ 0x80000000u) : ~k;
  return __uint_as_float(u);
}

// ---------------------------------------------------------------------------
// K0: init output + per-node softmax state
// ---------------------------------------------------------------------------
__global__ void gat_init(float* __restrict__ out, unsigned* __restrict__ seg_max_k,
                         float* __restrict__ seg_sum, int n_nodes, int out_elems) {
  int i = blockIdx.x * blockDim.x + threadIdx.x;
  int stride = gridDim.x * blockDim.x;
  for (int j = i; j < out_elems; j += stride) out[j] = 0.0f;
  for (int j = i; j < n_nodes; j += stride) {
    seg_max_k[j] = 0u;
    seg_sum[j] = 0.0f;
  }
}

// ---------------------------------------------------------------------------
// K1: node scores s_i = x@w_i, s_j = x@w_j via V_WMMA_F32_16X16X4_F32.
// One wave32 -> 16 nodes, 32 WMMAs over K=128.
// A 16x4 f32 layout: lanes 0-15 hold (K0,K1) of row M=lane; lanes 16-31 (K2,K3).
// B 4x16 f32 layout (symmetric): lane's column N = lane&15; lanes 0-15 supply
//   rows K0,K1, lanes 16-31 rows K2,K3.
// Key trick: D column n depends only on B column n; we read only D columns
// 0 (s_i) and 1 (s_j), so B columns >=2 may hold garbage. All lanes load
// unconditionally from (m==1 ? w_j : w_i) -> no divergence, EXEC stays full,
// straight-line global_load_b64 per operand.
// D 16x16 f32 layout: VGPR r: lanes 0-15 = row r, lanes 16-31 = row r+8.
// ---------------------------------------------------------------------------
__global__ void gat_node_scores(const float* __restrict__ x,
                                const float* __restrict__ w_i,
                                const float* __restrict__ w_j,
                                float* __restrict__ s_i,
                                float* __restrict__ s_j, int n_nodes) {
  const int lane  = threadIdx.x & 31;
  const int wave  = (blockIdx.x * blockDim.x + threadIdx.x) >> 5;
  const int node0 = wave * 16;
  if (node0 >= n_nodes) return;           // wave-uniform: EXEC all-1s for WMMA

  const int m     = lane & 15;            // row (for A) / column (for B)
  const int khalf = (lane >> 4) * 2;      // 0 for lanes 0-15, 2 for lanes 16-31
  const float* arow  = x + (size_t)(node0 + m) * HIDDEN + khalf;
  const float* wbase = (m == 1) ? w_j : w_i;   // cols >=2 load garbage (discarded)

  v8f c = {};
#pragma unroll
  for (int k = 0; k < HIDDEN; k += 4) {
    v2f a, b;
    a.x = arow[k + 0];
    a.y = arow[k + 1];
    b.x = wbase[k + khalf + 0];
    b.y = wbase[k + khalf + 1];
    c = __builtin_amdgcn_wmma_f32_16x16x4_f32(
        /*neg_a=*/false, a, /*neg_b=*/false, b,
        /*c_mod=*/(short)0, c, /*reuse_a=*/false, /*reuse_b=*/false);
  }

  // Column 0 of D = s_i (lanes 0,16), column 1 = s_j (lanes 1,17).
  const int mbase = node0 + (lane >> 4) * 8;
  if (m == 0) {
#pragma unroll
    for (int r = 0; r < 8; ++r) s_i[mbase + r] = c[r];
  } else if (m == 1) {
#pragma unroll
    for (int r = 0; r < 8; ++r) s_j[mbase + r] = c[r];
  }
}

// ---------------------------------------------------------------------------
// K2: e = leakyrelu(s_i[h] + s_j[t]); segment max over h via u32 atomicMax
// ---------------------------------------------------------------------------
__global__ void gat_edge_scores(const float* __restrict__ s_i, const float* __restrict__ s_j,
                                const int* __restrict__ h, const int* __restrict__ t,
                                float* __restrict__ e_buf, unsigned* __restrict__ seg_max_k,
                                int n_edges) {
  int i = blockIdx.x * blockDim.x + threadIdx.x;
  if (i >= n_edges) return;
  int hh = h[i];
  float e = s_i[hh] + s_j[t[i]];
  e = (e > 0.0f) ? e : LEAKY * e;
  e_buf[i] = e;
  atomicMax(&seg_max_k[hh], f32_key(e));
}

// ---------------------------------------------------------------------------
// K3: ex = exp(e - seg_max[h]); segment sum (in-place over e_buf)
// ---------------------------------------------------------------------------
__global__ void gat_edge_exp(float* __restrict__ e_buf, const unsigned* __restrict__ seg_max_k,
                             float* __restrict__ seg_sum, const int* __restrict__ h,
                             int n_edges) {
  int i = blockIdx.x * blockDim.x + threadIdx.x;
  if (i >= n_edges) return;
  int hh = h[i];
  float m  = key_f32(seg_max_k[hh]);
  float ex = __expf(e_buf[i] - m);
  e_buf[i] = ex;
  atomicAdd(&seg_sum[hh], ex);
}

// ---------------------------------------------------------------------------
// K4: scatter SpMM: out[h] += (ex/seg_sum[h]) * x[t]. One wave32 per edge,
// 4 dims per lane; coalesced b32 gathers + f32 atomics, all L2-resident.
// ---------------------------------------------------------------------------
__global__ void gat_spmm(const float* __restrict__ x, const float* __restrict__ ex_buf,
                         const float* __restrict__ seg_sum, const int* __restrict__ h,
                         const int* __restrict__ t, float* __restrict__ out, int n_edges) {
  const int lane = threadIdx.x & 31;
  const int edge = (blockIdx.x * blockDim.x + threadIdx.x) >> 5;
  if (edge >= n_edges) return;
  const int hh = h[edge], tt = t[edge];
  const float alpha = ex_buf[edge] / seg_sum[hh];
  const float* xr = x + (size_t)tt * HIDDEN;
  float* orow = out + (size_t)hh * HIDDEN;
#pragma unroll
  for (int d = lane; d < HIDDEN; d += 32)
    atomicAdd(&orow[d], alpha * xr[d]);
}

// ---------------------------------------------------------------------------
// K5: final ReLU
// ---------------------------------------------------------------------------
__global__ void gat_relu(float* __restrict__ out, int n) {
  int i = blockIdx.x * blockDim.x + threadIdx.x;
  int stride = gridDim.x * blockDim.x;
  for (int j = i; j < n; j += stride) out[j] = fmaxf(out[j], 0.0f);
}

// ---------------------------------------------------------------------------
extern "C" void kernel_launch(void* const* d_in, const int* in_sizes, int n_in,
                              void* d_out, int out_size, void* d_ws, size_t ws_size,
                              hipStream_t stream) {
  const float* x   = (const float*)d_in[0];
  const float* w_i = (const float*)d_in[1];
  const float* w_j = (const float*)d_in[2];
  const int*   h   = (const int*)d_in[3];
  const int*   t   = (const int*)d_in[4];
  float* out = (float*)d_out;

  const int n_nodes = in_sizes[0] / HIDDEN;
  const int n_edges = in_sizes[3];
  const int out_elems = n_nodes * HIDDEN;

  // workspace layout
  char* ws = (char*)d_ws;
  float*    s_i       = (float*)ws;    ws += (size_t)n_nodes * 4;
  float*    s_j       = (float*)ws;    ws += (size_t)n_nodes * 4;
  unsigned* seg_max_k = (unsigned*)ws; ws += (size_t)n_nodes * 4;
  float*    seg_sum   = (float*)ws;    ws += (size_t)n_nodes * 4;
  float*    e_buf     = (float*)ws;    // n_edges floats (reused as ex)

  const int TB = 256;

  // K0: init
  gat_init<<<2048, TB, 0, stream>>>(out, seg_max_k, seg_sum, n_nodes, out_elems);

  // K1: WMMA node scores (16 nodes / wave, 8 waves / block)
  int waves = (n_nodes + 15) / 16;
  int blocks1 = (waves + 7) / 8;
  gat_node_scores<<<blocks1, TB, 0, stream>>>(x, w_i, w_j, s_i, s_j, n_nodes);

  // K2: edge scores + segment max
  gat_edge_scores<<<(n_edges + TB - 1) / TB, TB, 0, stream>>>(s_i, s_j, h, t, e_buf,
                                                              seg_max_k, n_edges);

  // K3: exp + segment sum
  gat_edge_exp<<<(n_edges + TB - 1) / TB, TB, 0, stream>>>(e_buf, seg_max_k, seg_sum, h,
                                                           n_edges);

  // K4: scatter SpMM (1 edge per wave, 8 edges per block)
  int blocks4 = (n_edges + 7) / 8;
  gat_spmm<<<blocks4, TB, 0, stream>>>(x, e_buf, seg_sum, h, t, out, n_edges);

  // K5: ReLU
  gat_relu<<<2048, TB, 0, stream>>>(out, out_elems);
}